// SelfAttentionConv2d_1580547968133
// MI455X (gfx1250) — compile-verified
//
#include <hip/hip_runtime.h>
#include <hip/hip_bf16.h>

#define B_    4
#define CIN_  256
#define COUT_ 256
#define H_    48
#define W_    48
#define S_    (H_ * W_)      // 2304 interior pixels per image
#define NPIX  (B_ * S_)      // 9216
#define KH_   7
#define KW_   7

typedef float v2f __attribute__((ext_vector_type(2)));
typedef float v8f __attribute__((ext_vector_type(8)));

// ---------------------------------------------------------------------------
// Kernel 1: q/k/v projections as fp32 WMMA GEMMs.
//   Y[b][s][o] = sum_c Wm[o][c] * x[b][c][s]   (interior pixels only; the
//   zero-pad ring of the reference produces exactly-zero k/v there, handled
//   as OOB-zero in kernel 2).
// One wave per 16(M=cout) x 16(N=pixels) tile, K=256 via 64 x V_WMMA_F32_16X16X4_F32,
// split into two independent accumulator chains for ILP.
// Output stored PIXEL-MAJOR [b*S+s][cout] for coalesced reads in kernel 2.
// ---------------------------------------------------------------------------
__global__ __launch_bounds__(32) void proj_wmma_f32(
    const float* __restrict__ x,
    const float* __restrict__ wq, const float* __restrict__ wk,
    const float* __restrict__ wv,
    float* __restrict__ qb, float* __restrict__ kb, float* __restrict__ vb)
{
    const int lane = threadIdx.x & 31;
    const int lm   = lane & 15;     // row (A) / col (B,D) within tile
    const int half = lane >> 4;     // K sub-group / D row offset

    const int ntile = blockIdx.x;   // 0..575  (pixel tiles)
    const int mtile = blockIdx.y;   // 0..15   (cout tiles)
    const int which = blockIdx.z;   // 0:q 1:k 2:v

    const float* Wm = (which == 0) ? wq : (which == 1) ? wk : wv;
    float*       Y  = (which == 0) ? qb : (which == 1) ? kb : vb;

    const int n0 = ntile * 16;          // 2304 % 16 == 0 -> tile stays in one image
    const int b  = n0 / S_;
    const int s0 = n0 - b * S_;

    const float* xb   = x  + (size_t)b * CIN_ * S_ + s0;          // xb[c*S_ + lm]
    const float* wrow = Wm + (size_t)(mtile * 16 + lm) * CIN_;    // A row per lane

    v8f acc0 = {};
    v8f acc1 = {};
    #pragma unroll 8
    for (int k0 = 0; k0 < CIN_; k0 += 8) {
        // chain 0: K = k0 .. k0+3
        {
            const int kk = k0 + half * 2;
            v2f a = *(const v2f*)(wrow + kk);               // A[m][kk], A[m][kk+1]
            v2f bm;
            bm.x = xb[(size_t)(kk    ) * S_ + lm];          // B[kk  ][n]
            bm.y = xb[(size_t)(kk + 1) * S_ + lm];          // B[kk+1][n]
            acc0 = __builtin_amdgcn_wmma_f32_16x16x4_f32(
                       false, a, false, bm, (short)0, acc0, false, false);
        }
        // chain 1: K = k0+4 .. k0+7
        {
            const int kk = k0 + 4 + half * 2;
            v2f a = *(const v2f*)(wrow + kk);
            v2f bm;
            bm.x = xb[(size_t)(kk    ) * S_ + lm];
            bm.y = xb[(size_t)(kk + 1) * S_ + lm];
            acc1 = __builtin_amdgcn_wmma_f32_16x16x4_f32(
                       false, a, false, bm, (short)0, acc1, false, false);
        }
    }

    // D layout: reg r -> row (mtile*16 + r + 8*half), col (n0 + lm); pixel-major store
    float* yb = Y + (size_t)(b * S_ + s0 + lm) * COUT_ + mtile * 16 + half * 8;
    #pragma unroll
    for (int r = 0; r < 8; ++r) yb[r] = acc0[r] + acc1[r];
}

// ---------------------------------------------------------------------------
// Kernel 2: per-pixel windowed attention. One wave32 per output pixel.
// Lane owns channels c = lane + 32j (j=0..7) and taps {lane, lane+32}.
// ---------------------------------------------------------------------------
__global__ __launch_bounds__(256) void attn_win_kernel(
    const float* __restrict__ qb, const float* __restrict__ kb,
    const float* __restrict__ vb,
    const float* __restrict__ rel_x, const float* __restrict__ rel_y,
    const float* __restrict__ bias,  float* __restrict__ out)
{
    const int lane = threadIdx.x & 31;
    const int wave = threadIdx.x >> 5;
    const int pix  = blockIdx.x * 8 + wave;      // 0..9215
    const int b = pix / S_;
    const int s = pix - b * S_;
    const int h = s / W_;
    const int w = s - h * W_;

    // q for this pixel, distributed: qr[j] = q[c = lane+32j]
    const float* qp = qb + (size_t)pix * COUT_;
    float qr[8];
    #pragma unroll
    for (int j = 0; j < 8; ++j) qr[j] = qp[lane + 32 * j];

    // qx[l] = sum_{c<128} q[c]*rel_x[c][l]; qy[k] = sum_{c>=128} q[c]*rel_y[c-128][k]
    // full xor-reductions leave the value replicated in every lane.
    float qxv[KW_], qyv[KH_];
    #pragma unroll
    for (int l = 0; l < KW_; ++l) {
        float p = 0.f;
        #pragma unroll
        for (int j = 0; j < 4; ++j) p += qr[j] * rel_x[(lane + 32 * j) * KW_ + l];
        for (int off = 16; off; off >>= 1) p += __shfl_xor(p, off, 32);
        qxv[l] = p;
    }
    #pragma unroll
    for (int kk = 0; kk < KH_; ++kk) {
        float p = 0.f;
        #pragma unroll
        for (int j = 0; j < 4; ++j) p += qr[4 + j] * rel_y[(lane + 32 * j) * KH_ + kk];
        for (int off = 16; off; off >>= 1) p += __shfl_xor(p, off, 32);
        qyv[kk] = p;
    }

    // scores over the 49 taps; OOB taps contribute dot=0 but still get qy+qx
    float scA = 0.f, scB = 0.f;                  // taps lane and lane+32
    for (int t = 0; t < 49; ++t) {
        const int dk = t / 7, dl = t - 7 * dk;
        const int hh = h + dk - 3, ww = w + dl - 3;
        float p = 0.f;
        if (hh >= 0 && hh < H_ && ww >= 0 && ww < W_) {
            const float* kp = kb + (size_t)(b * S_ + hh * W_ + ww) * COUT_;
            #pragma unroll
            for (int j = 0; j < 8; ++j) p += qr[j] * kp[lane + 32 * j];
        }
        for (int off = 16; off; off >>= 1) p += __shfl_xor(p, off, 32);
        p += qyv[dk] + qxv[dl];
        if (t < 32) { if (lane == t)      scA = p; }
        else        { if (lane == t - 32) scB = p; }
    }

    // softmax over 49 (lane<17 also owns tap lane+32)
    const bool hasB = (lane < 17);
    float m = hasB ? fmaxf(scA, scB) : scA;
    for (int off = 16; off; off >>= 1) m = fmaxf(m, __shfl_xor(m, off, 32));
    const float eA = expf(scA - m);
    const float eB = hasB ? expf(scB - m) : 0.f;
    float ssum = eA + eB;
    for (int off = 16; off; off >>= 1) ssum += __shfl_xor(ssum, off, 32);
    const float inv = 1.f / ssum;
    const float aA = eA * inv, aB = eB * inv;

    // out[c] = sum_t attn[t] * v[c][tap t]
    float acc[8] = {0.f, 0.f, 0.f, 0.f, 0.f, 0.f, 0.f, 0.f};
    for (int t = 0; t < 49; ++t) {
        const float at = (t < 32) ? __shfl(aA, t, 32) : __shfl(aB, t - 32, 32);
        const int dk = t / 7, dl = t - 7 * dk;
        const int hh = h + dk - 3, ww = w + dl - 3;
        if (hh >= 0 && hh < H_ && ww >= 0 && ww < W_) {
            const float* vp = vb + (size_t)(b * S_ + hh * W_ + ww) * COUT_;
            #pragma unroll
            for (int j = 0; j < 8; ++j) acc[j] += at * vp[lane + 32 * j];
        }
    }

    float* op = out + (size_t)b * COUT_ * S_ + s;     // out[b][c][h][w]
    #pragma unroll
    for (int j = 0; j < 8; ++j) {
        const int c = lane + 32 * j;
        op[(size_t)c * S_] = acc[j] + bias[c];
    }
}

extern "C" void kernel_launch(void* const* d_in, const int* in_sizes, int n_in,
                              void* d_out, int out_size, void* d_ws, size_t ws_size,
                              hipStream_t stream) {
    const float* x     = (const float*)d_in[0];
    const float* w_q   = (const float*)d_in[1];
    const float* w_k   = (const float*)d_in[2];
    const float* w_v   = (const float*)d_in[3];
    const float* bias  = (const float*)d_in[4];
    const float* rel_x = (const float*)d_in[5];
    const float* rel_y = (const float*)d_in[6];
    float* outp = (float*)d_out;

    // workspace: q,k,v pixel-major, each NPIX*COUT floats (3 * 9.4 MB)
    float* qbuf = (float*)d_ws;
    float* kbuf = qbuf + (size_t)NPIX * COUT_;
    float* vbuf = kbuf + (size_t)NPIX * COUT_;

    dim3 gProj(NPIX / 16, COUT_ / 16, 3);   // (576, 16, 3), one wave per 16x16 tile
    proj_wmma_f32<<<gProj, 32, 0, stream>>>(x, w_q, w_k, w_v, qbuf, kbuf, vbuf);

    attn_win_kernel<<<NPIX / 8, 256, 0, stream>>>(qbuf, kbuf, vbuf,
                                                  rel_x, rel_y, bias, outp);
}